// TimeMixingOptimized_16071767621910
// MI455X (gfx1250) — compile-verified
//
#include <hip/hip_runtime.h>
#include <hip/hip_bf16.h>
#include <math.h>

// Problem constants (from the reference: B=4, T=4096, C=1024)
#define Bsz 4
#define Tsz 4096
#define Csz 1024
#define Msz (Bsz * Tsz)          // 16384 rows
#define SCHUNKS 64               // scan chunks per (b,c) chain
#define LCHUNK  (Tsz / SCHUNKS)  // 64 timesteps per chunk

typedef __attribute__((ext_vector_type(2))) float v2f;
typedef __attribute__((ext_vector_type(4))) float v4f;
typedef __attribute__((ext_vector_type(8))) float v8f;

// D(16x16 f32) = A(16x4 f32) x B(4x16 f32) + C  -> v_wmma_f32_16x16x4_f32
static __device__ inline v8f wmma4_f32(v2f a, v2f b, v8f c) {
    return __builtin_amdgcn_wmma_f32_16x16x4_f32(
        /*neg_a=*/false, a, /*neg_b=*/false, b,
        /*c_mod=*/(short)0, c, /*reuse_a=*/false, /*reuse_b=*/false);
}

// ---------------------------------------------------------------------------
// Kernel 1: fused time-shift + token mixing + 3 GEMMs (K = kin*Wk^T, etc.)
// One wave per 32(M) x 64(N) strip: 2 M-tiles x 4 N-tiles x 3 outputs
// = 24 f32 WMMA accumulators. M-blocking amortizes the dominant weight
// fragment loads (12 per K-step shared by both M-tiles).
// A-fragment layout (16x4 f32): lane = {row=lane&15, half=lane>>4};
//   VGPR0 = A[row][2*half], VGPR1 = A[row][2*half+1].
// B-fragment (4x16): lane holds B[2*half + {0,1}][row] = W[n0+row][k + 2*half + {0,1}].
// ---------------------------------------------------------------------------
__global__ __launch_bounds__(32)
void rwkv_qkv_wmma(const float* __restrict__ x,
                   const float* __restrict__ Wk,
                   const float* __restrict__ Wv,
                   const float* __restrict__ Wr,
                   const float* __restrict__ mixk,
                   const float* __restrict__ mixv,
                   const float* __restrict__ mixr,
                   float* __restrict__ Kout,
                   float* __restrict__ Vout,
                   float* __restrict__ Rout)
{
    const int lane = threadIdx.x & 31;
    const int row  = lane & 15;
    const int half = lane >> 4;
    const int n0   = blockIdx.x * 64;   // N strip (4 tiles of 16)
    const int m0   = blockIdx.y * 32;   // M strip (2 tiles of 16)
    const int mA   = m0 + row;
    const int mB   = m0 + 16 + row;
    // t==0 (prev_x = 0) can only occur for rows with m % 32 == 0, i.e. only
    // in the first M-tile at row 0; the second tile never needs the predicate.
    const bool zprevA = ((mA % Tsz) == 0);

    const float* xrA = x + (size_t)mA * Csz;
    const float* xpA = x + ((size_t)mA - 1) * Csz;  // only dereferenced when !zprevA
    const float* xrB = x + (size_t)mB * Csz;
    const float* xpB = x + ((size_t)mB - 1) * Csz;

    v8f accK0[4] = {}; v8f accK1[4] = {};
    v8f accV0[4] = {}; v8f accV1[4] = {};
    v8f accR0[4] = {}; v8f accR1[4] = {};

    for (int k0 = 0; k0 < Csz; k0 += 4) {
        const int kk = k0 + half * 2;
        v2f xv0 = *(const v2f*)(xrA + kk);
        v2f pv0 = {0.0f, 0.0f};
        if (!zprevA) pv0 = *(const v2f*)(xpA + kk);
        v2f xv1 = *(const v2f*)(xrB + kk);
        v2f pv1 = *(const v2f*)(xpB + kk);
        v2f mk = *(const v2f*)(mixk + kk);
        v2f mv = *(const v2f*)(mixv + kk);
        v2f mr = *(const v2f*)(mixr + kk);
        v2f d0 = xv0 - pv0;
        v2f d1 = xv1 - pv1;
        v2f aK0 = pv0 + mk * d0;   // x*mix + prev*(1-mix)
        v2f aV0 = pv0 + mv * d0;
        v2f aR0 = pv0 + mr * d0;
        v2f aK1 = pv1 + mk * d1;
        v2f aV1 = pv1 + mv * d1;
        v2f aR1 = pv1 + mr * d1;
        #pragma unroll
        for (int j = 0; j < 4; ++j) {
            const size_t woff = (size_t)(n0 + j * 16 + row) * Csz + kk;
            v2f bK = *(const v2f*)(Wk + woff);
            v2f bV = *(const v2f*)(Wv + woff);
            v2f bR = *(const v2f*)(Wr + woff);
            accK0[j] = wmma4_f32(aK0, bK, accK0[j]);
            accK1[j] = wmma4_f32(aK1, bK, accK1[j]);
            accV0[j] = wmma4_f32(aV0, bV, accV0[j]);
            accV1[j] = wmma4_f32(aV1, bV, accV1[j]);
            accR0[j] = wmma4_f32(aR0, bR, accR0[j]);
            accR1[j] = wmma4_f32(aR1, bR, accR1[j]);
        }
    }

    // C/D layout: VGPR i -> row (mtile + half*8 + i), col (n0 + j*16 + row)
    #pragma unroll
    for (int j = 0; j < 4; ++j) {
        #pragma unroll
        for (int i = 0; i < 8; ++i) {
            const size_t offA = (size_t)(m0 + half * 8 + i) * Csz + (n0 + j * 16 + row);
            const size_t offB = (size_t)(m0 + 16 + half * 8 + i) * Csz + (n0 + j * 16 + row);
            Kout[offA] = accK0[j][i];  Kout[offB] = accK1[j][i];
            Vout[offA] = accV0[j][i];  Vout[offB] = accV1[j][i];
            Rout[offA] = accR0[j][i];  Rout[offB] = accR1[j][i];
        }
    }
}

// ---------------------------------------------------------------------------
// WKV scan, segmented into SCHUNKS chunks of LCHUNK steps for parallelism.
// Representation: actual N = num * e^m (log-max stabilized), N_t = e^w N_{t-1} + e^{k_t} v_t.
// ---------------------------------------------------------------------------

// Pass 1: per-chunk partial sums starting from zero state.
__global__ __launch_bounds__(256)
void wkv_pass1(const float* __restrict__ K, const float* __restrict__ V,
               const float* __restrict__ time_decay,
               float* __restrict__ sumN, float* __restrict__ sumD, float* __restrict__ sumM)
{
    const int tid = blockIdx.x * blockDim.x + threadIdx.x;   // B*C*S threads
    const int c = tid % Csz;
    const int s = (tid / Csz) % SCHUNKS;
    const int b = tid / (Csz * SCHUNKS);
    const float w = -__expf(time_decay[c]);

    float num = 0.0f, den = 0.0f, m = -1e38f;
    size_t base = ((size_t)b * Tsz + (size_t)s * LCHUNK) * Csz + c;
    for (int t = 0; t < LCHUNK; ++t) {
        const float kk = K[base];
        const float vv = V[base];
        const float ms = fmaxf(m + w, kk);
        const float e1 = __expf(m + w - ms);
        const float e2 = __expf(kk - ms);
        num = e1 * num + e2 * vv;
        den = e1 * den + e2;
        m = ms;
        base += Csz;
    }
    const int o = (b * SCHUNKS + s) * Csz + c;
    sumN[o] = num; sumD[o] = den; sumM[o] = m;
}

// Pass 2: sequential prefix-combine over chunk summaries (per (b,c) chain).
// state_in(s+1) = e^{w*L} * state_in(s)  +  C_s
__global__ __launch_bounds__(256)
void wkv_pass2(const float* __restrict__ time_decay,
               const float* __restrict__ sumN, const float* __restrict__ sumD,
               const float* __restrict__ sumM,
               float* __restrict__ inN, float* __restrict__ inD, float* __restrict__ inM)
{
    const int tid = blockIdx.x * blockDim.x + threadIdx.x;   // B*C threads
    const int c = tid % Csz;
    const int b = tid / Csz;
    const float wL = -__expf(time_decay[c]) * (float)LCHUNK;

    float N = 0.0f, D = 0.0f, M = -1e38f;
    for (int s = 0; s < SCHUNKS; ++s) {
        const int o = (b * SCHUNKS + s) * Csz + c;
        inN[o] = N; inD[o] = D; inM[o] = M;
        const float ns = sumN[o], ds = sumD[o], ms = sumM[o];
        const float M2 = fmaxf(M + wL, ms);
        const float e1 = __expf(M + wL - M2);
        const float e2 = __expf(ms - M2);
        N = e1 * N + e2 * ns;
        D = e1 * D + e2 * ds;
        M = M2;
    }
}

// Pass 3: replay each chunk from its correct incoming state, emit outputs.
// wkv output overwrites K in place (k[t] is dead once read).
__global__ __launch_bounds__(256)
void wkv_pass3(float* __restrict__ K, const float* __restrict__ V,
               const float* __restrict__ time_decay, const float* __restrict__ time_first,
               const float* __restrict__ inN, const float* __restrict__ inD,
               const float* __restrict__ inM)
{
    const int tid = blockIdx.x * blockDim.x + threadIdx.x;   // B*C*S threads
    const int c = tid % Csz;
    const int s = (tid / Csz) % SCHUNKS;
    const int b = tid / (Csz * SCHUNKS);
    const float w = -__expf(time_decay[c]);
    const float u = time_first[c];

    const int o = (b * SCHUNKS + s) * Csz + c;
    float num = inN[o], den = inD[o], m = inM[o];

    size_t base = ((size_t)b * Tsz + (size_t)s * LCHUNK) * Csz + c;
    for (int t = 0; t < LCHUNK; ++t) {
        const float kk = K[base];
        const float vv = V[base];
        // output (uses u bonus, state from t-1)
        const float mo = fmaxf(m, kk + u);
        const float e1 = __expf(m - mo);
        const float e2 = __expf(kk + u - mo);
        K[base] = (e1 * num + e2 * vv) / (e1 * den + e2);
        // state update with decay w
        const float ms  = fmaxf(m + w, kk);
        const float e1s = __expf(m + w - ms);
        const float e2s = __expf(kk - ms);
        num = e1s * num + e2s * vv;
        den = e1s * den + e2s;
        m = ms;
        base += Csz;
    }
}

// rwkv = sigmoid(R) * wkv, in place into R (float4 vectorized, pure-BW pass).
__global__ __launch_bounds__(256)
void rwkv_sigmul(float* __restrict__ R, const float* __restrict__ wkv, int n4)
{
    const int i = blockIdx.x * blockDim.x + threadIdx.x;
    if (i < n4) {
        v4f r = *(const v4f*)(R + 4 * (size_t)i);
        v4f wv = *(const v4f*)(wkv + 4 * (size_t)i);
        v4f o;
        #pragma unroll
        for (int e = 0; e < 4; ++e)
            o[e] = wv[e] / (1.0f + __expf(-r[e]));
        *(v4f*)(R + 4 * (size_t)i) = o;
    }
}

// Output GEMM: out = rwkv @ Wo^T. One wave per 32(M) x 64(N) strip, 8 accums.
__global__ __launch_bounds__(32)
void rwkv_out_wmma(const float* __restrict__ A, const float* __restrict__ Wo,
                   float* __restrict__ out)
{
    const int lane = threadIdx.x & 31;
    const int row  = lane & 15;
    const int half = lane >> 4;
    const int n0   = blockIdx.x * 64;
    const int m0   = blockIdx.y * 32;

    const float* arA = A + (size_t)(m0 + row) * Csz;
    const float* arB = A + (size_t)(m0 + 16 + row) * Csz;
    v8f acc0[4] = {};
    v8f acc1[4] = {};

    for (int k0 = 0; k0 < Csz; k0 += 4) {
        const int kk = k0 + half * 2;
        v2f a0 = *(const v2f*)(arA + kk);
        v2f a1 = *(const v2f*)(arB + kk);
        #pragma unroll
        for (int j = 0; j < 4; ++j) {
            v2f bb = *(const v2f*)(Wo + (size_t)(n0 + j * 16 + row) * Csz + kk);
            acc0[j] = wmma4_f32(a0, bb, acc0[j]);
            acc1[j] = wmma4_f32(a1, bb, acc1[j]);
        }
    }

    #pragma unroll
    for (int j = 0; j < 4; ++j) {
        #pragma unroll
        for (int i = 0; i < 8; ++i) {
            out[(size_t)(m0 + half * 8 + i) * Csz + (n0 + j * 16 + row)] = acc0[j][i];
            out[(size_t)(m0 + 16 + half * 8 + i) * Csz + (n0 + j * 16 + row)] = acc1[j][i];
        }
    }
}

extern "C" void kernel_launch(void* const* d_in, const int* in_sizes, int n_in,
                              void* d_out, int out_size, void* d_ws, size_t ws_size,
                              hipStream_t stream) {
    (void)in_sizes; (void)n_in; (void)out_size; (void)ws_size;

    const float* x    = (const float*)d_in[0];
    const float* Wk   = (const float*)d_in[1];
    const float* Wv   = (const float*)d_in[2];
    const float* Wr   = (const float*)d_in[3];
    const float* Wo   = (const float*)d_in[4];
    const float* td   = (const float*)d_in[5];   // time_decay [C]
    const float* tf   = (const float*)d_in[6];   // time_first [C]
    const float* mixk = (const float*)d_in[7];
    const float* mixv = (const float*)d_in[8];
    const float* mixr = (const float*)d_in[9];
    float* out = (float*)d_out;

    // Workspace layout (~198 MB): K | V | R | chunk summaries | incoming states
    const size_t MC = (size_t)Msz * Csz;              // 16.7M floats each
    const size_t SC = (size_t)Bsz * SCHUNKS * Csz;    // 262144 floats each
    float* Kbuf = (float*)d_ws;
    float* Vbuf = Kbuf + MC;
    float* Rbuf = Vbuf + MC;
    float* sumN = Rbuf + MC;
    float* sumD = sumN + SC;
    float* sumM = sumD + SC;
    float* inN  = sumM + SC;
    float* inD  = inN + SC;
    float* inM  = inD + SC;

    // 1) fused mixing + K/V/R GEMMs (WMMA f32), 32Mx64N per wave
    dim3 gGemm(Csz / 64, Msz / 32);   // (16, 512) waves of 32
    rwkv_qkv_wmma<<<gGemm, 32, 0, stream>>>(x, Wk, Wv, Wr, mixk, mixv, mixr,
                                            Kbuf, Vbuf, Rbuf);

    // 2) segmented WKV scan (3 passes); wkv overwrites Kbuf
    const int nChunkThreads = Bsz * Csz * SCHUNKS;    // 262144
    wkv_pass1<<<nChunkThreads / 256, 256, 0, stream>>>(Kbuf, Vbuf, td, sumN, sumD, sumM);
    wkv_pass2<<<(Bsz * Csz) / 256, 256, 0, stream>>>(td, sumN, sumD, sumM, inN, inD, inM);
    wkv_pass3<<<nChunkThreads / 256, 256, 0, stream>>>(Kbuf, Vbuf, td, tf, inN, inD, inM);

    // 3) rwkv = sigmoid(R) * wkv (in place into Rbuf), float4 per thread
    rwkv_sigmul<<<(int)(MC / 4 / 256), 256, 0, stream>>>(Rbuf, Kbuf, (int)(MC / 4));

    // 4) output GEMM, 32Mx64N per wave
    rwkv_out_wmma<<<gGemm, 32, 0, stream>>>(Rbuf, Wo, out);
}